// ECNN_56384330662355
// MI455X (gfx1250) — compile-verified
//
#include <hip/hip_runtime.h>
#include <hip/hip_bf16.h>

typedef float v2f __attribute__((ext_vector_type(2)));
typedef float v8f __attribute__((ext_vector_type(8)));

#define MN 1024
#define WAVES 4
#define TPB (WAVES * 32)
#define TSTRIDE 68   // 16x64 tile row stride (padded; keeps 16B alignment, avoids column bank conflicts)

__device__ __forceinline__ float silu_f(float v) {
    // v * rcp(1 + e^-v): v_exp + v_rcp + muls, much cheaper than precise fdiv
    return v * __builtin_amdgcn_rcpf(1.0f + __expf(-v));
}

// ---------------- Kernel 1: pe_i = h @ We1[:64], pe_j = h @ We1[64:128] ----------------
__global__ __launch_bounds__(256) void pe_kernel(const float* __restrict__ h,
                                                 const float* __restrict__ We1,
                                                 float* __restrict__ peI,
                                                 float* __restrict__ peJ) {
    int idx = blockIdx.x * 256 + threadIdx.x;   // 0..131071
    int which = idx >> 16;                      // 0 -> peI, 1 -> peJ
    int rem = idx & 65535;
    int node = rem >> 6;
    int k = rem & 63;
    const float* W = We1 + which * (64 * 64);
    const float* hrow = h + node * 64;
    float acc = 0.f;
#pragma unroll 8
    for (int f = 0; f < 64; ++f) acc += hrow[f] * W[f * 64 + k];
    (which ? peJ : peI)[node * 64 + k] = acc;
}

// ---------------- Kernel 2: fused edge MLPs (WMMA f32 16x16x4) + reductions ----------------
// B fragments are kernel-invariant -> staged via swizzled LDS, then hoisted into
// registers (16 kt x 4 nt x v2f x 2 matrices = 256 VGPRs) so the GEMM inner loops
// are pure WMMA streams with only A-side LDS traffic.
__global__ __launch_bounds__(TPB) void edge_kernel(
    const float* __restrict__ x, const float* __restrict__ a,
    const float* __restrict__ We1, const float* __restrict__ be1,
    const float* __restrict__ We2, const float* __restrict__ be2,
    const float* __restrict__ Wx1, const float* __restrict__ bx1,
    const float* __restrict__ Wx2, const float* __restrict__ bx2,
    const float* __restrict__ peI, const float* __restrict__ peJ,
    float* __restrict__ maccOut, float* __restrict__ xaccOut)
{
    __shared__ __align__(16) float sWe2f[64 * 64];
    __shared__ __align__(16) float sWx1f[64 * 64];
    __shared__ __align__(16) float sPeiB[64], sW1d2[64], sW1a[64];
    __shared__ float sBe2[64], sBx1[64], sWx2v[64];
    __shared__ __align__(16) float tbuf[WAVES][16 * TSTRIDE];
    __shared__ float sdx[WAVES][16], sdy[WAVES][16], sdz[WAVES][16], sS[WAVES][16];
    __shared__ float rMacc[WAVES][64];
    __shared__ float rX[WAVES * 16][4];

    const int i    = blockIdx.x;
    const int tid  = threadIdx.x;
    const int wv   = tid >> 5;
    const int lane = tid & 31;
    const int l16  = lane & 15;
    const bool hi  = (lane >= 16);
    const int koff  = hi ? 2 : 0;   // A/B lane-half K offset
    const int Mbase = hi ? 8 : 0;   // C/D lane-half M offset

    // stage weights (B-fragment swizzled) + vectors
    for (int t = tid; t < 4096; t += TPB) {
        int frag   = t >> 6;          // 0..63
        int within = t & 63;
        int ln     = within >> 1;     // source lane 0..31
        int idx    = within & 1;
        int kt     = frag >> 2;
        int nt     = frag & 3;
        int k      = kt * 4 + ((ln < 16) ? 0 : 2) + idx;
        int n      = nt * 16 + (ln & 15);
        sWe2f[t] = We2[k * 64 + n];
        sWx1f[t] = Wx1[k * 64 + n];
    }
    if (tid < 64) {
        sPeiB[tid] = peI[i * 64 + tid] + be1[tid];   // fold be1 into pe_i
        sBe2[tid]  = be2[tid];
        sBx1[tid]  = bx1[tid];
        sWx2v[tid] = Wx2[tid];
        sW1d2[tid] = We1[128 * 64 + tid];   // d2 row of We1
        sW1a[tid]  = We1[129 * 64 + tid];   // a  row of We1
    }
    __syncthreads();

    // hoist all B fragments into registers (kernel-invariant)
    v2f bwe[16][4], bwx[16][4];
#pragma unroll
    for (int kt = 0; kt < 16; ++kt) {
#pragma unroll
        for (int nt = 0; nt < 4; ++nt) {
            bwe[kt][nt] = *(const v2f*)&sWe2f[(kt * 4 + nt) * 64 + lane * 2];
            bwx[kt][nt] = *(const v2f*)&sWx1f[(kt * 4 + nt) * 64 + lane * 2];
        }
    }

    const float xi0 = x[3 * i], xi1 = x[3 * i + 1], xi2 = x[3 * i + 2];
    const float bx2v = bx2[0];

    float macc4[4] = {0.f, 0.f, 0.f, 0.f};
    float xaccx = 0.f, xaccy = 0.f, xaccz = 0.f;

    for (int tt = 0; tt < 16; ++tt) {
        const int j0 = (tt * WAVES + wv) * 16;   // 16-edge tile: (i, j0..j0+15)
        const int excl = i - j0;                  // masked row if in [0,16)

        // ---- build t = silu(pre): each lane owns row l16, columns [c0, c0+32) ----
        {
            const int j = j0 + l16;
            const float dx = xi0 - x[3 * j];
            const float dy = xi1 - x[3 * j + 1];
            const float dz = xi2 - x[3 * j + 2];
            const float d2v = dx * dx + dy * dy + dz * dz;
            const float avv = a[(size_t)i * MN + j];
            if (lane < 16) { sdx[wv][lane] = dx; sdy[wv][lane] = dy; sdz[wv][lane] = dz; }
            const int c0 = hi ? 32 : 0;
            const float* pjrow = peJ + (size_t)j * 64;
#pragma unroll
            for (int cc = 0; cc < 32; cc += 4) {
                const int c = c0 + cc;
                float4 pj = *(const float4*)&pjrow[c];
                float4 pi = *(const float4*)&sPeiB[c];
                float4 wd = *(const float4*)&sW1d2[c];
                float4 wa = *(const float4*)&sW1a[c];
                float4 tv;
                tv.x = silu_f(pi.x + pj.x + d2v * wd.x + avv * wa.x);
                tv.y = silu_f(pi.y + pj.y + d2v * wd.y + avv * wa.y);
                tv.z = silu_f(pi.z + pj.z + d2v * wd.z + avv * wa.z);
                tv.w = silu_f(pi.w + pj.w + d2v * wd.w + avv * wa.w);
                *(float4*)&tbuf[wv][l16 * TSTRIDE + c] = tv;
            }
        }
        __builtin_amdgcn_wave_barrier();

        // ---- GEMM1: c1 = t @ We2 + be2 ----
        v8f c1[4];
#pragma unroll
        for (int nt = 0; nt < 4; ++nt) {
            float bval = sBe2[nt * 16 + l16];
#pragma unroll
            for (int r = 0; r < 8; ++r) c1[nt][r] = bval;
        }
#pragma unroll
        for (int kt = 0; kt < 16; ++kt) {
            v2f av = *(const v2f*)&tbuf[wv][l16 * TSTRIDE + kt * 4 + koff];
#pragma unroll
            for (int nt = 0; nt < 4; ++nt) {
                c1[nt] = __builtin_amdgcn_wmma_f32_16x16x4_f32(
                    false, av, false, bwe[kt][nt], (short)0, c1[nt], false, false);
            }
        }
        __builtin_amdgcn_wave_barrier();

        // m = silu(c1): accumulate masked row-sum, write back for GEMM2 A-reads
#pragma unroll
        for (int nt = 0; nt < 4; ++nt) {
#pragma unroll
            for (int r = 0; r < 8; ++r) {
                float mv = silu_f(c1[nt][r]);
                int Mr = Mbase + r;
                if (Mr != excl) macc4[nt] += mv;
                tbuf[wv][Mr * TSTRIDE + nt * 16 + l16] = mv;
            }
        }
        __builtin_amdgcn_wave_barrier();

        // ---- GEMM2: c2 = m @ Wx1 + bx1 ----
        v8f c2[4];
#pragma unroll
        for (int nt = 0; nt < 4; ++nt) {
            float bval = sBx1[nt * 16 + l16];
#pragma unroll
            for (int r = 0; r < 8; ++r) c2[nt][r] = bval;
        }
#pragma unroll
        for (int kt = 0; kt < 16; ++kt) {
            v2f av = *(const v2f*)&tbuf[wv][l16 * TSTRIDE + kt * 4 + koff];
#pragma unroll
            for (int nt = 0; nt < 4; ++nt) {
                c2[nt] = __builtin_amdgcn_wmma_f32_16x16x4_f32(
                    false, av, false, bwx[kt][nt], (short)0, c2[nt], false, false);
            }
        }

        // s-partials: p[r] = sum_N silu(u) * Wx2[N]
        float p[8] = {0.f, 0.f, 0.f, 0.f, 0.f, 0.f, 0.f, 0.f};
#pragma unroll
        for (int nt = 0; nt < 4; ++nt) {
            float wxv = sWx2v[nt * 16 + l16];
#pragma unroll
            for (int r = 0; r < 8; ++r) p[r] += silu_f(c2[nt][r]) * wxv;
        }
        // reduce p across the 16-lane half (masks < 16 stay within half)
#pragma unroll
        for (int msk = 1; msk < 16; msk <<= 1) {
#pragma unroll
            for (int r = 0; r < 8; ++r) p[r] += __shfl_xor(p[r], msk, 32);
        }
        if (lane < 8)                      sS[wv][lane]      = p[lane];        // M = 0..7
        else if (lane >= 16 && lane < 24)  sS[wv][lane - 8]  = p[lane - 16];   // M = 8..15
        __builtin_amdgcn_wave_barrier();

        // coordinate accumulation (mask j == i)
        if (lane < 16) {
            float sv = sS[wv][lane] + bx2v;
            if (j0 + lane != i) {
                xaccx += sdx[wv][lane] * sv;
                xaccy += sdy[wv][lane] * sv;
                xaccz += sdz[wv][lane] * sv;
            }
        }
        __builtin_amdgcn_wave_barrier();
    }

    // combine M-halves of the message accumulator, then block-level reduce
#pragma unroll
    for (int nt = 0; nt < 4; ++nt) macc4[nt] += __shfl_xor(macc4[nt], 16, 32);
    if (lane < 16) {
#pragma unroll
        for (int nt = 0; nt < 4; ++nt) rMacc[wv][nt * 16 + lane] = macc4[nt];
        rX[wv * 16 + lane][0] = xaccx;
        rX[wv * 16 + lane][1] = xaccy;
        rX[wv * 16 + lane][2] = xaccz;
    }
    __syncthreads();
    if (tid < 64) {
        float s = 0.f;
        for (int w = 0; w < WAVES; ++w) s += rMacc[w][tid];
        maccOut[(size_t)i * 64 + tid] = s;
    }
    if (tid < 3) {
        float s = 0.f;
        for (int e = 0; e < WAVES * 16; ++e) s += rX[e][tid];
        xaccOut[(size_t)i * 3 + tid] = s;
    }
}

// ---------------- Kernel 3: phi_h node update + coordinate finalize ----------------
__global__ __launch_bounds__(64) void node_kernel(
    const float* __restrict__ x, const float* __restrict__ h,
    const float* __restrict__ Wh1, const float* __restrict__ bh1,
    const float* __restrict__ Wh2, const float* __restrict__ bh2,
    const float* __restrict__ macc, const float* __restrict__ xacc,
    float* __restrict__ xout, float* __restrict__ hout)
{
    __shared__ float sh[64], sm[64], g[64];
    const int i = blockIdx.x, k = threadIdx.x;
    sh[k] = h[i * 64 + k];
    sm[k] = macc[i * 64 + k];
    __syncthreads();
    float acc = bh1[k];
#pragma unroll 8
    for (int f = 0; f < 64; ++f) acc += sh[f] * Wh1[f * 64 + k];
#pragma unroll 8
    for (int f = 0; f < 64; ++f) acc += sm[f] * Wh1[(64 + f) * 64 + k];
    g[k] = silu_f(acc);
    __syncthreads();
    float o = bh2[k];
#pragma unroll 8
    for (int f = 0; f < 64; ++f) o += g[f] * Wh2[f * 64 + k];
    hout[i * 64 + k] = o;
    if (k < 3) xout[i * 3 + k] = x[i * 3 + k] + (1.0f / 1023.0f) * xacc[i * 3 + k];
}

extern "C" void kernel_launch(void* const* d_in, const int* in_sizes, int n_in,
                              void* d_out, int out_size, void* d_ws, size_t ws_size,
                              hipStream_t stream) {
    const float* x   = (const float*)d_in[0];
    const float* a   = (const float*)d_in[1];
    const float* h   = (const float*)d_in[2];
    const float* We1 = (const float*)d_in[3];
    const float* be1 = (const float*)d_in[4];
    const float* We2 = (const float*)d_in[5];
    const float* be2 = (const float*)d_in[6];
    const float* Wx1 = (const float*)d_in[7];
    const float* bx1 = (const float*)d_in[8];
    const float* Wx2 = (const float*)d_in[9];
    const float* bx2 = (const float*)d_in[10];
    const float* Wh1 = (const float*)d_in[11];
    const float* bh1 = (const float*)d_in[12];
    const float* Wh2 = (const float*)d_in[13];
    const float* bh2 = (const float*)d_in[14];

    float* ws      = (float*)d_ws;
    float* peI     = ws;                // 1024*64
    float* peJ     = ws + 65536;        // 1024*64
    float* maccBuf = ws + 131072;       // 1024*64
    float* xaccBuf = ws + 196608;       // 1024*3

    float* xout = (float*)d_out;        // 1024*3
    float* hout = (float*)d_out + 3072; // 1024*64

    pe_kernel<<<512, 256, 0, stream>>>(h, We1, peI, peJ);
    edge_kernel<<<MN, TPB, 0, stream>>>(x, a, We1, be1, We2, be2, Wx1, bx1,
                                        Wx2, bx2, peI, peJ, maccBuf, xaccBuf);
    node_kernel<<<MN, 64, 0, stream>>>(x, h, Wh1, bh1, Wh2, bh2,
                                       maccBuf, xaccBuf, xout, hout);
}